// Yolov4Decoder_39350490366622
// MI455X (gfx1250) — compile-verified
//
#include <hip/hip_runtime.h>
#include <hip/hip_bf16.h>

// ---------------- problem constants ----------------
#define NB        16          // batch
#define NANCH     29772       // 76*76*3 + 52*52*3 + 38*38*3
#define NPAD      32768       // next pow2 for bitonic sort
#define CHUNK     4096        // LDS-resident sort chunk (32KB)
#define TOPN      1000
#define MAXOBJ    100
#define MIN_SCORE 0.05f
#define NMS_THR   0.5f

// workspace layout (bytes)
#define OFF_KEYS   0ull
#define SZ_KEYS    ((size_t)NB * NPAD * 8)            // 4,194,304
#define OFF_SCORES (OFF_KEYS + SZ_KEYS)
#define SZ_SCORES  ((size_t)NB * NANCH * 4)           // 1,905,408
#define OFF_CLS    (OFF_SCORES + SZ_SCORES)
#define SZ_CLS     ((size_t)NB * NANCH * 4)
#define OFF_BOXES  (OFF_CLS + SZ_CLS)                 // 16B aligned

// vector/pointer types for the gfx1250 async global<->LDS builtins
typedef int v2i __attribute__((ext_vector_type(2)));
typedef __attribute__((address_space(1))) v2i* gptr_v2i;
typedef __attribute__((address_space(3))) v2i* lptr_v2i;

#if defined(__HIP_DEVICE_COMPILE__) && __has_builtin(__builtin_amdgcn_global_load_async_to_lds_b64)
#define HAVE_ASYNC_LOAD 1
#else
#define HAVE_ASYNC_LOAD 0
#endif
#if defined(__HIP_DEVICE_COMPILE__) && __has_builtin(__builtin_amdgcn_global_store_async_from_lds_b64)
#define HAVE_ASYNC_STORE 1
#else
#define HAVE_ASYNC_STORE 0
#endif

__device__ __forceinline__ void wait_async0() {
#if defined(__HIP_DEVICE_COMPILE__)
#if __has_builtin(__builtin_amdgcn_s_wait_asynccnt)
    __builtin_amdgcn_s_wait_asynccnt(0);
#else
    asm volatile("s_wait_asynccnt 0" ::: "memory");
#endif
#endif
}

__device__ __forceinline__ void ldsLoad64(unsigned long long* dstLds,
                                          const unsigned long long* srcGlobal) {
#if HAVE_ASYNC_LOAD
    __builtin_amdgcn_global_load_async_to_lds_b64((gptr_v2i)srcGlobal, (lptr_v2i)dstLds, 0, 0);
#else
    *dstLds = *srcGlobal;
#endif
}

__device__ __forceinline__ void ldsStore64(unsigned long long* dstGlobal,
                                           const unsigned long long* srcLds) {
#if HAVE_ASYNC_STORE
    __builtin_amdgcn_global_store_async_from_lds_b64((gptr_v2i)dstGlobal, (lptr_v2i)srcLds, 0, 0);
#else
    *dstGlobal = *srcLds;
#endif
}

// ordered-uint mapping of float: monotone, so descending uint == descending float
__device__ __forceinline__ unsigned int orderKey(float f) {
    unsigned int u = __float_as_uint(f);
    return (u & 0x80000000u) ? ~u : (u | 0x80000000u);
}

// ---------------- decode: one thread per anchor ----------------
__global__ void decode_level(const float* __restrict__ head,
                             const float* __restrict__ anch,
                             unsigned long long* __restrict__ keys,
                             float* __restrict__ scores,
                             int*   __restrict__ cls,
                             float* __restrict__ boxes,
                             int nPerImg, int levelOff) {
    int t = blockIdx.x * blockDim.x + threadIdx.x;
    int total = NB * nPerImg;
    if (t >= total) return;
    int img = t / nPerImg;
    int a   = t - img * nPerImg;

    const float* h = head + (size_t)t * 85;
    // streaming one-shot read of 162MB: non-temporal so we don't thrash L2
    float obj = __builtin_nontemporal_load(h + 0);
    float r0 = __builtin_nontemporal_load(h + 1);
    float r1 = __builtin_nontemporal_load(h + 2);
    float r2 = __builtin_nontemporal_load(h + 3);
    float r3 = __builtin_nontemporal_load(h + 4);

    // first-occurrence argmax over 80 classes (strict >)
    float best = __builtin_nontemporal_load(h + 5);
    int bc = 0;
#pragma unroll 4
    for (int c = 1; c < 80; ++c) {
        float v = __builtin_nontemporal_load(h + 5 + c);
        if (v > best) { best = v; bc = c; }
    }
    float sc = best * obj;

    const float* an = anch + (size_t)t * 5;
    float ax = an[0], ay = an[1], aw = an[2], ah = an[3], st = an[4];
    float cx = (r0 + ax) * st;
    float cy = (r1 + ay) * st;
    float w  = r2 * aw * st;
    float hh = r3 * ah * st;
    float x1 = truncf(cx - w * 0.5f);
    float y1 = truncf(cy - hh * 0.5f);
    float x2 = truncf(cx + w * 0.5f);
    float y2 = truncf(cy + hh * 0.5f);

    int gi = levelOff + a;
    size_t gidx = (size_t)img * NANCH + gi;
    scores[gidx] = sc;
    cls[gidx]    = bc;
    float* bp = boxes + gidx * 4;
    bp[0] = x1; bp[1] = y1; bp[2] = x2; bp[3] = y2;

    float masked = (sc > MIN_SCORE) ? sc : -__builtin_inff();
    unsigned int k32 = orderKey(masked);
    // stable: score desc, then index asc (~gi makes smaller gi sort first under "descending")
    keys[((size_t)img << 15) + gi] =
        ((unsigned long long)k32 << 32) | (unsigned int)(~(unsigned int)gi);
}

// ---------------- phase 1: sort each 4096-chunk fully in LDS (k = 2..4096) ----------------
__global__ __launch_bounds__(1024)
void sort_local(unsigned long long* __restrict__ keys) {
    __shared__ unsigned long long sk[CHUNK];
    const int tid   = threadIdx.x;
    const int img   = blockIdx.x >> 3;       // 8 chunks per image
    const int chunk = blockIdx.x & 7;
    const int base  = chunk << 12;           // element index within image
    unsigned long long* g = keys + ((size_t)img << 15) + base;

    for (int e = tid; e < CHUNK; e += 1024) ldsLoad64(&sk[e], g + e);
    wait_async0();
    __syncthreads();
    // substitute key 0 for pad entries (sorts after everything incl. -inf scores)
    for (int e = tid; e < CHUNK; e += 1024)
        if (base + e >= NANCH) sk[e] = 0ull;
    __syncthreads();

    for (int k = 2; k <= CHUNK; k <<= 1) {
        for (int j = k >> 1; j > 0; j >>= 1) {
            for (int p = tid; p < (CHUNK / 2); p += 1024) {
                int i = ((p & ~(j - 1)) << 1) | (p & (j - 1));
                int l = i + j;
                unsigned long long a = sk[i];
                unsigned long long b = sk[l];
                bool up = (((base + i) & k) == 0);
                bool sw = up ? (a < b) : (a > b);       // descending overall
                if (sw) { sk[i] = b; sk[l] = a; }
            }
            __syncthreads();
        }
    }

    for (int e = tid; e < CHUNK; e += 1024) ldsStore64(g + e, &sk[e]);
    wait_async0();
}

// ---------------- global bitonic stage (only spans j >= 4096) ----------------
__global__ void bitonic_global(unsigned long long* __restrict__ keys, int k, int j) {
    unsigned int p   = blockIdx.x * blockDim.x + threadIdx.x;   // NB * 16384 pairs
    unsigned int img = p >> 14;
    unsigned int pp  = p & 16383u;
    unsigned int i   = ((pp & ~(unsigned int)(j - 1)) << 1) | (pp & (unsigned int)(j - 1));
    unsigned int l   = i + (unsigned int)j;
    unsigned long long* base = keys + ((size_t)img << 15);
    unsigned long long a = base[i];
    unsigned long long b = base[l];
    bool up = ((i & (unsigned int)k) == 0);
    bool sw = up ? (a < b) : (a > b);
    if (sw) { base[i] = b; base[l] = a; }
}

// ---------------- local merge tail: j = 2048..1 for a given global k ----------------
__global__ __launch_bounds__(1024)
void merge_local(unsigned long long* __restrict__ keys, int kglob) {
    __shared__ unsigned long long sk[CHUNK];
    const int tid   = threadIdx.x;
    const int img   = blockIdx.x >> 3;
    const int chunk = blockIdx.x & 7;
    const int base  = chunk << 12;
    unsigned long long* g = keys + ((size_t)img << 15) + base;

    for (int e = tid; e < CHUNK; e += 1024) ldsLoad64(&sk[e], g + e);
    wait_async0();
    __syncthreads();

    for (int j = CHUNK / 2 / 2 * 2; j > 0; j >>= 1) {   // j = 2048..1
        for (int p = tid; p < (CHUNK / 2); p += 1024) {
            int i = ((p & ~(j - 1)) << 1) | (p & (j - 1));
            int l = i + j;
            unsigned long long a = sk[i];
            unsigned long long b = sk[l];
            bool up = (((base + i) & kglob) == 0);
            bool sw = up ? (a < b) : (a > b);
            if (sw) { sk[i] = b; sk[l] = a; }
        }
        __syncthreads();
    }

    for (int e = tid; e < CHUNK; e += 1024) ldsStore64(g + e, &sk[e]);
    wait_async0();
}

// ---------------- NMS + compaction + output, one block per image ----------------
__global__ __launch_bounds__(1024)
void nms_out(const unsigned long long* __restrict__ keys,
             const float* __restrict__ scores,
             const int*   __restrict__ cls,
             const float* __restrict__ boxes,
             float* __restrict__ out) {
    __shared__ unsigned long long skey[TOPN];
    __shared__ float bx[TOPN][4];
    __shared__ float sarea[TOPN];
    __shared__ float ssc[TOPN];
    __shared__ int   scl[TOPN];
    __shared__ int   ssup[TOPN];
    __shared__ int   spos[TOPN];
    __shared__ float o_s[MAXOBJ];
    __shared__ float o_c[MAXOBJ];
    __shared__ float o_b[MAXOBJ * 4];

    const int tid = threadIdx.x;
    const int img = blockIdx.x;
    const unsigned long long* gk = keys + ((size_t)img << 15);

    // stage top-1000 sorted keys into LDS via the async path
    if (tid < TOPN) ldsLoad64(&skey[tid], gk + tid);
    wait_async0();
    __syncthreads();

    // gather per-entry data (top-1000 are always real anchors: pads sort last)
    if (tid < TOPN) {
        unsigned long long kk = skey[tid];
        unsigned int idx = ~(unsigned int)(kk & 0xFFFFFFFFull);
        size_t g = (size_t)img * NANCH + idx;
        float sc = scores[g];
        ssc[tid] = sc;
        scl[tid] = cls[g];
        const float* bp = boxes + g * 4;
        float x1 = bp[0], y1 = bp[1], x2 = bp[2], y2 = bp[3];
        bx[tid][0] = x1; bx[tid][1] = y1; bx[tid][2] = x2; bx[tid][3] = y2;
        float w = x2 - x1, h = y2 - y1;
        sarea[tid] = fmaxf(w * h, 1e-4f);
        ssup[tid]  = (sc > MIN_SCORE) ? 0 : 1;     // supp init = !valid
    }
    __syncthreads();

    // greedy NMS (exact reference semantics)
    for (int i = 0; i < TOPN; ++i) {
        bool keep_i = (ssup[i] == 0) && (ssc[i] > MIN_SCORE);   // block-uniform
        if (keep_i) {
            float ix1 = bx[i][0], iy1 = bx[i][1], ix2 = bx[i][2], iy2 = bx[i][3];
            float ia  = sarea[i];
            if (tid > i && tid < TOPN) {
                float tlx = fmaxf(ix1, bx[tid][0]);
                float tly = fmaxf(iy1, bx[tid][1]);
                float brx = fminf(ix2, bx[tid][2]);
                float bry = fminf(iy2, bx[tid][3]);
                float ow = fmaxf(brx - tlx, 0.0f);
                float oh = fmaxf(bry - tly, 0.0f);
                float ov = ow * oh;
                float un = fmaxf(ia + sarea[tid] - ov, 1e-4f);
                if (ov / un >= NMS_THR) ssup[tid] = 1;
            }
            __syncthreads();
        }
    }
    __syncthreads();

    // serial prefix scan of keep flags
    if (tid == 0) {
        int p = 0;
        for (int i = 0; i < TOPN; ++i) {
            int kflag = (ssup[i] == 0) && (ssc[i] > MIN_SCORE);
            spos[i] = kflag ? p : -1;
            p += kflag;
        }
    }
    __syncthreads();

    // defaults, scatter, write out
    if (tid < MAXOBJ) { o_s[tid] = -1.0f; o_c[tid] = -1.0f; }
    if (tid < MAXOBJ * 4) o_b[tid] = 0.0f;
    __syncthreads();
    if (tid < TOPN) {
        int p = spos[tid];
        if (p >= 0 && p < MAXOBJ) {
            o_s[p] = ssc[tid];
            o_c[p] = (float)scl[tid];
            o_b[p * 4 + 0] = bx[tid][0];
            o_b[p * 4 + 1] = bx[tid][1];
            o_b[p * 4 + 2] = bx[tid][2];
            o_b[p * 4 + 3] = bx[tid][3];
        }
    }
    __syncthreads();
    if (tid < MAXOBJ) {
        out[img * MAXOBJ + tid]               = o_s[tid];   // scores  [16,100]
        out[NB * MAXOBJ + img * MAXOBJ + tid] = o_c[tid];   // classes [16,100]
    }
    if (tid < MAXOBJ * 4) {
        out[2 * NB * MAXOBJ + img * MAXOBJ * 4 + tid] = o_b[tid];  // boxes [16,100,4]
    }
}

// ---------------- host launcher ----------------
extern "C" void kernel_launch(void* const* d_in, const int* in_sizes, int n_in,
                              void* d_out, int out_size, void* d_ws, size_t ws_size,
                              hipStream_t stream) {
    // setup_inputs() dict order: head0, anchors0, head1, anchors1, head2, anchors2
    const float* head0 = (const float*)d_in[0];
    const float* anch0 = (const float*)d_in[1];
    const float* head1 = (const float*)d_in[2];
    const float* anch1 = (const float*)d_in[3];
    const float* head2 = (const float*)d_in[4];
    const float* anch2 = (const float*)d_in[5];
    float* out = (float*)d_out;

    char* w = (char*)d_ws;
    unsigned long long* keys = (unsigned long long*)(w + OFF_KEYS);
    float* scores = (float*)(w + OFF_SCORES);
    int*   cls    = (int*)(w + OFF_CLS);
    float* boxes  = (float*)(w + OFF_BOXES);

    // 1) decode all three levels (bandwidth-bound: ~170MB streamed)
    {
        int n0 = 76 * 76 * 3;   // 17328
        int n1 = 52 * 52 * 3;   //  8112
        int n2 = 38 * 38 * 3;   //  4332
        decode_level<<<(NB * n0 + 255) / 256, 256, 0, stream>>>(head0, anch0, keys, scores, cls, boxes, n0, 0);
        decode_level<<<(NB * n1 + 255) / 256, 256, 0, stream>>>(head1, anch1, keys, scores, cls, boxes, n1, n0);
        decode_level<<<(NB * n2 + 255) / 256, 256, 0, stream>>>(head2, anch2, keys, scores, cls, boxes, n2, n0 + n1);
    }

    // 2) hierarchical bitonic sort, descending, 32768 keys per image:
    //    - sort_local: all stages k=2..4096 in LDS (also zero-fills pad keys)
    //    - bitonic_global: only the 6 stages with span j>=4096
    //    - merge_local: j=2048..1 tails in LDS
    {
        const int locBlocks  = NB * (NPAD / CHUNK);   // 128
        const int glbBlocks  = (NB * (NPAD / 2)) / 256;  // 1024 (one thread per pair)
        sort_local<<<locBlocks, 1024, 0, stream>>>(keys);
        bitonic_global<<<glbBlocks, 256, 0, stream>>>(keys, 8192, 4096);
        merge_local<<<locBlocks, 1024, 0, stream>>>(keys, 8192);
        bitonic_global<<<glbBlocks, 256, 0, stream>>>(keys, 16384, 8192);
        bitonic_global<<<glbBlocks, 256, 0, stream>>>(keys, 16384, 4096);
        merge_local<<<locBlocks, 1024, 0, stream>>>(keys, 16384);
        bitonic_global<<<glbBlocks, 256, 0, stream>>>(keys, 32768, 16384);
        bitonic_global<<<glbBlocks, 256, 0, stream>>>(keys, 32768, 8192);
        bitonic_global<<<glbBlocks, 256, 0, stream>>>(keys, 32768, 4096);
        merge_local<<<locBlocks, 1024, 0, stream>>>(keys, 32768);
    }

    // 3) NMS + compaction + output (one block per image)
    nms_out<<<NB, 1024, 0, stream>>>(keys, scores, cls, boxes, out);
}